// GELU13_17566416240645
// MI455X (gfx1250) — compile-verified
//
#include <hip/hip_runtime.h>
#include <math.h>

typedef __attribute__((ext_vector_type(2))) float v2f;
typedef __attribute__((ext_vector_type(8))) float v8f;

#define DIM   2048
#define NCLUS 8
#define NROWS 16384
#define MOM   0.999f

__device__ __forceinline__ float wave_red_add(float v) {
#pragma unroll
  for (int off = 16; off >= 1; off >>= 1) v += __shfl_xor(v, off, 32);
  return v;
}

// ---------------------------------------------------------------- zero scratch
__global__ void kZero(float* __restrict__ sums, int* __restrict__ counts,
                      float* __restrict__ Ppad, float* __restrict__ P2pad) {
  const int i = blockIdx.x * 256 + threadIdx.x;
  if (i < NCLUS * DIM) sums[i] = 0.f;
  if (i < NCLUS) counts[i] = 0;
  if (i < 16 * DIM) { Ppad[i] = 0.f; P2pad[i] = 0.f; }
}

// ------------------------------------------------- P_norm into padded buffer
__global__ void kPnorm(const float* __restrict__ protos, float* __restrict__ Ppad) {
  const int lane = threadIdx.x & 31;
  const int k    = threadIdx.x >> 5;   // 8 waves = 8 clusters
  float ss = 0.f;
  for (int c = lane; c < DIM; c += 32) {
    const float v = protos[k * DIM + c];
    ss = fmaf(v, v, ss);
  }
  ss = wave_red_add(ss);
  const float inv = 1.f / fmaxf(sqrtf(ss), 1e-12f);
  for (int c = lane; c < DIM; c += 32)
    Ppad[k * DIM + c] = protos[k * DIM + c] * inv;
}

// ---- fused pass 1: WMMA dots vs P_norm -> argmax -> LDS segment-sum
// Each WG owns 128 rows: 8 waves x one 16-row WMMA tile each.
// Phase 2 re-reads the same 128 rows (L2/WGP$-hot) into an 8x2048 LDS
// accumulator, then flushes once with global atomics.
__global__ void kAssignSum(const float* __restrict__ x, const float* __restrict__ Ppad,
                           float* __restrict__ inv_xn, int* __restrict__ counts,
                           float* __restrict__ sums) {
  extern __shared__ char smem[];
  float* acc     = (float*)smem;                        // NCLUS*DIM floats (64 KB)
  int*   lassign = (int*)(smem + NCLUS * DIM * 4);      // 128 ints
  int*   lcounts = lassign + 128;                       // NCLUS ints

  const int tid = threadIdx.x;
  for (int i = tid; i < NCLUS * DIM; i += 256) acc[i] = 0.f;
  if (tid < NCLUS) lcounts[tid] = 0;
  __syncthreads();

  const int lane = tid & 31;
  const int wid  = tid >> 5;
  const int h    = lane >> 4;   // half of wave
  const int m    = lane & 15;   // row (A) / cluster col (B,C)
  const size_t base = ((size_t)blockIdx.x * 8 + wid) * 16;

  const float2* __restrict__ xr = (const float2*)(x    + (base + (size_t)m) * DIM) + h;
  const float2* __restrict__ pr = (const float2*)(Ppad + (size_t)m * DIM) + h;

  v8f c = {};
  float ss = 0.f;
#pragma unroll 8
  for (int k0 = 0; k0 < DIM; k0 += 4) {
    const float2 av = xr[k0 >> 1];        // A[m][k0+2h], A[m][k0+2h+1]
    const float2 bv = pr[k0 >> 1];        // B fragment from padded P rows
    v2f a; a.x = av.x; a.y = av.y;
    v2f b; b.x = bv.x; b.y = bv.y;
    ss = fmaf(av.x, av.x, fmaf(av.y, av.y, ss));
    c = __builtin_amdgcn_wmma_f32_16x16x4_f32(false, a, false, b, (short)0, c,
                                              false, false);
  }
  // row sum-of-squares: combine halves (k-phase 0/1 vs 2/3)
  const float ssrow = ss + __shfl_xor(ss, 16, 32);
  if (h == 0) inv_xn[base + m] = 1.f / fmaxf(sqrtf(ssrow), 1e-8f);

  // argmax over clusters (argmax invariant under positive per-row scaling)
#pragma unroll
  for (int v = 0; v < 8; ++v) {
    float val = c[v];                     // dot(row v+8h, cluster m)
    int   idx = m;
    if (m >= NCLUS) { val = -3.0e38f; idx = 0; }  // padded clusters
#pragma unroll
    for (int off = 4; off >= 1; off >>= 1) {      // reduce within group of 8
      const float oval = __shfl_xor(val, off, 32);
      const int   oidx = __shfl_xor(idx, off, 32);
      if (oval > val || (oval == val && oidx < idx)) { val = oval; idx = oidx; }
    }
    if (m == 0) {                         // lanes 0 and 16
      lassign[wid * 16 + v + 8 * h] = idx;
      atomicAdd(&lcounts[idx], 1);
    }
  }
  __syncthreads();

  // phase 2: per-WG segment sums; each thread owns 8 columns for all clusters
  const size_t rbase = (size_t)blockIdx.x * 128;
  const int c0 = tid * 8;
  for (int r = 0; r < 128; ++r) {
    const int a = lassign[r];             // uniform across block
    const float4* __restrict__ xv = (const float4*)(x + (rbase + r) * (size_t)DIM);
    const float4 v0 = xv[tid * 2 + 0];
    const float4 v1 = xv[tid * 2 + 1];
    float4* dst = (float4*)&acc[a * DIM + c0];
    float4 d0 = dst[0], d1 = dst[1];
    d0.x += v0.x; d0.y += v0.y; d0.z += v0.z; d0.w += v0.w;
    d1.x += v1.x; d1.y += v1.y; d1.z += v1.z; d1.w += v1.w;
    dst[0] = d0; dst[1] = d1;
  }
  __syncthreads();
  for (int i = tid; i < NCLUS * DIM; i += 256) atomicAdd(&sums[i], acc[i]);
  if (tid < NCLUS) atomicAdd(&counts[tid], lcounts[tid]);
}

// ---------------- centroids -> EMA -> P2n (padded)
__global__ void kP2(const float* __restrict__ protos, const float* __restrict__ sums,
                    const int* __restrict__ counts, float* __restrict__ P2pad) {
  const int lane = threadIdx.x & 31;
  const int k    = threadIdx.x >> 5;
  const float cnt = (float)counts[k];
  const bool  has = cnt > 0.f;
  const float rden = 1.f / fmaxf(cnt, 1.f);
  const float* __restrict__ srow = sums   + k * DIM;
  const float* __restrict__ prow = protos + k * DIM;

  float ss = 0.f;
  for (int c = lane; c < DIM; c += 32) {
    const float cv = has ? srow[c] * rden : prow[c];
    ss = fmaf(cv, cv, ss);
  }
  ss = wave_red_add(ss);
  const float invc = 1.f / fmaxf(sqrtf(ss), 1e-12f);

  float ss2 = 0.f;
  for (int c = lane; c < DIM; c += 32) {
    const float cv = has ? srow[c] * rden : prow[c];
    const float p2 = MOM * prow[c] + (1.f - MOM) * (cv * invc);
    ss2 = fmaf(p2, p2, ss2);
  }
  ss2 = wave_red_add(ss2);
  const float inv2 = 1.f / fmaxf(sqrtf(ss2), 1e-8f);

  for (int c = lane; c < DIM; c += 32) {
    const float cv = has ? srow[c] * rden : prow[c];
    const float p2 = MOM * prow[c] + (1.f - MOM) * (cv * invc);
    P2pad[k * DIM + c] = p2 * inv2;
  }
}

// ---------------- pass 3: WMMA dots vs P2n, novelty scale, fused GELU
__global__ void kFinal(const float* __restrict__ x, const float* __restrict__ P2pad,
                       const float* __restrict__ inv_xn,
                       const float* __restrict__ log_tau,
                       const float* __restrict__ log_blend,
                       float* __restrict__ out) {
  __shared__ float sscale[8 * 16];
  const int tid  = threadIdx.x;
  const int lane = tid & 31;
  const int wid  = tid >> 5;
  const int h    = lane >> 4;
  const int m    = lane & 15;
  const size_t base = ((size_t)blockIdx.x * 8 + wid) * 16;

  const float2* __restrict__ xr = (const float2*)(x     + (base + (size_t)m) * DIM) + h;
  const float2* __restrict__ pr = (const float2*)(P2pad + (size_t)m * DIM) + h;

  v8f c = {};
#pragma unroll 8
  for (int k0 = 0; k0 < DIM; k0 += 4) {
    const float2 av = xr[k0 >> 1];
    const float2 bv = pr[k0 >> 1];
    v2f a; a.x = av.x; a.y = av.y;
    v2f b; b.x = bv.x; b.y = bv.y;
    c = __builtin_amdgcn_wmma_f32_16x16x4_f32(false, a, false, b, (short)0, c,
                                              false, false);
  }

  const float tau   = expf(log_tau[0]);
  const float alpha = 1.f / (1.f + expf(-log_blend[0]));

#pragma unroll
  for (int v = 0; v < 8; ++v) {
    const float inv = inv_xn[base + 8 * h + v];     // row of this C fragment
    float val = fminf(fmaxf(c[v] * inv, -1.f), 1.f);
    if (m >= NCLUS) val = -3.0e38f;
#pragma unroll
    for (int off = 4; off >= 1; off >>= 1)
      val = fmaxf(val, __shfl_xor(val, off, 32));
    if (m == 0) {
      const float dist  = fminf(fmaxf(1.f - val, 0.f), 2.f);
      const float nov   = 1.f - expf(-tau * dist);
      const float scale = fminf(fmaxf(1.f - alpha + alpha * nov, 0.1f), 10.f);
      sscale[wid * 16 + v + 8 * h] = scale;
    }
  }
  __syncthreads();

  const float CG = 0.7978845608028654f;  // sqrt(2/pi)
  for (int r = 0; r < 16; ++r) {
    const float s = sscale[wid * 16 + r];
    const size_t row = base + r;
    const float4* __restrict__ xv = (const float4*)(x   + row * (size_t)DIM);
    float4*       __restrict__ ov = (float4*)      (out + row * (size_t)DIM);
#pragma unroll 4
    for (int it = 0; it < 16; ++it) {   // 2048 cols / (32 lanes * 4)
      float4 t = xv[it * 32 + lane];
      t.x *= s; t.y *= s; t.z *= s; t.w *= s;
      float4 g;
      g.x = 0.5f * t.x * (1.f + tanhf(CG * fmaf(0.044715f * t.x, t.x * t.x, t.x)));
      g.y = 0.5f * t.y * (1.f + tanhf(CG * fmaf(0.044715f * t.y, t.y * t.y, t.y)));
      g.z = 0.5f * t.z * (1.f + tanhf(CG * fmaf(0.044715f * t.z, t.z * t.z, t.z)));
      g.w = 0.5f * t.w * (1.f + tanhf(CG * fmaf(0.044715f * t.w, t.w * t.w, t.w)));
      ov[it * 32 + lane] = g;
    }
  }
}

// ----------------------------------------------------------------- launcher
extern "C" void kernel_launch(void* const* d_in, const int* in_sizes, int n_in,
                              void* d_out, int out_size, void* d_ws, size_t ws_size,
                              hipStream_t stream) {
  const float* x         = (const float*)d_in[0];
  const float* protos    = (const float*)d_in[1];
  const float* log_tau   = (const float*)d_in[2];
  const float* log_blend = (const float*)d_in[3];
  float* out = (float*)d_out;

  // workspace layout (~390 KB)
  float* Ppad   = (float*)d_ws;             // 16*DIM (rows 8..15 zero)
  float* P2pad  = Ppad  + 16 * DIM;         // 16*DIM
  float* sums   = P2pad + 16 * DIM;         // NCLUS*DIM
  float* inv_xn = sums  + NCLUS * DIM;      // NROWS
  int*   counts = (int*)(inv_xn + NROWS);   // NCLUS

  const size_t smem = (size_t)NCLUS * DIM * 4 + 128 * 4 + NCLUS * 4; // 66080 B
  kZero     <<<128, 256, 0,    stream>>>(sums, counts, Ppad, P2pad);
  kPnorm    <<<1,   256, 0,    stream>>>(protos, Ppad);
  kAssignSum<<<128, 256, smem, stream>>>(x, Ppad, inv_xn, counts, sums);
  kP2       <<<1,   256, 0,    stream>>>(protos, sums, counts, P2pad);
  kFinal    <<<128, 256, 0,    stream>>>(x, P2pad, inv_xn, log_tau, log_blend, out);
}